// GGCNLSPELayer_46961172414535
// MI455X (gfx1250) — compile-verified
//
#include <hip/hip_runtime.h>
#include <math.h>

#define NN 50000
#define EE 800000
#define D  64

typedef __attribute__((ext_vector_type(2))) float v2f;
typedef __attribute__((ext_vector_type(8))) float v8f;

// ---------------- Fused node projections ----------------
// blockIdx.y in [0,24): slab = (output o in [0,6)) * 4 + (col tile nt in [0,4))
// Each wave holds the B fragments (weights) for its slab in registers and
// grid-strides over 16-row tiles, streaming A and issuing back-to-back WMMAs.
__global__ void node_gemm_kernel(
    const float* __restrict__ h, const float* __restrict__ p,
    const float* __restrict__ WA1, const float* __restrict__ bA1,
    const float* __restrict__ WA2, const float* __restrict__ bA2,
    const float* __restrict__ WB1, const float* __restrict__ bB1,
    const float* __restrict__ WB2, const float* __restrict__ bB2,
    const float* __restrict__ WC1, const float* __restrict__ bC1,
    const float* __restrict__ WC2, const float* __restrict__ bC2,
    float* __restrict__ A1h, float* __restrict__ A2hp,
    float* __restrict__ B1h, float* __restrict__ B2h,
    float* __restrict__ C1p, float* __restrict__ C2p) {
  const int lane = threadIdx.x & 31;
  const int wave = threadIdx.x >> 5;
  const int half = lane >> 4;
  const int r    = lane & 15;
  const int o    = blockIdx.y >> 2;     // which output
  const int nt   = blockIdx.y & 3;      // which 16-col tile
  const int col  = nt * 16 + r;

  const float* W;  const float* bias;  float* out;
  const float* in0; const float* in1 = nullptr;
  switch (o) {
    case 0:  W = WA1; bias = bA1; out = A1h;  in0 = h; in1 = p; break;
    case 1:  W = WA2; bias = bA2; out = A2hp; in0 = h; in1 = p; break;
    case 2:  W = WB1; bias = bB1; out = B1h;  in0 = h; break;
    case 3:  W = WB2; bias = bB2; out = B2h;  in0 = h; break;
    case 4:  W = WC1; bias = bC1; out = C1p;  in0 = p; break;
    default: W = WC2; bias = bC2; out = C2p;  in0 = p; break;
  }
  const bool k128 = (o < 2);

  // Register-resident B fragments (loop-invariant)
  v2f bf[32];
#pragma unroll
  for (int ks = 0; ks < 16; ++ks) {
    const int k = ks * 4 + 2 * half;
    bf[ks].x = W[(k + 0) * D + col];
    bf[ks].y = W[(k + 1) * D + col];
  }
  if (k128) {
#pragma unroll
    for (int ks = 0; ks < 16; ++ks) {
      const int k = 64 + ks * 4 + 2 * half;
      bf[16 + ks].x = W[(k + 0) * D + col];
      bf[16 + ks].y = W[(k + 1) * D + col];
    }
  }
  const float bcol = bias[col];

  const int tiles   = NN / 16;
  const int wstride = gridDim.x * 8;
  for (int tile = blockIdx.x * 8 + wave; tile < tiles; tile += wstride) {
    const int row0 = tile * 16;
    const float* arow = in0 + (size_t)(row0 + r) * D;
    v2f af[16];
#pragma unroll
    for (int ks = 0; ks < 16; ++ks) {
      const int k = ks * 4 + 2 * half;
      af[ks].x = arow[k]; af[ks].y = arow[k + 1];
    }
    v8f c = {};
#pragma unroll
    for (int ks = 0; ks < 16; ++ks)
      c = __builtin_amdgcn_wmma_f32_16x16x4_f32(false, af[ks], false, bf[ks],
                                                (short)0, c, false, false);
    if (k128) {
      const float* arow1 = in1 + (size_t)(row0 + r) * D;
      v2f ag[16];
#pragma unroll
      for (int ks = 0; ks < 16; ++ks) {
        const int k = ks * 4 + 2 * half;
        ag[ks].x = arow1[k]; ag[ks].y = arow1[k + 1];
      }
#pragma unroll
      for (int ks = 0; ks < 16; ++ks)
        c = __builtin_amdgcn_wmma_f32_16x16x4_f32(false, ag[ks], false,
                                                  bf[16 + ks], (short)0, c,
                                                  false, false);
    }
#pragma unroll
    for (int i = 0; i < 8; ++i)
      out[(size_t)(row0 + i + 8 * half) * D + col] = c[i] + bcol;
  }
}

// -------- Edge GEMM + gather + sigmoid + segment-sum of sigma --------
// blockIdx.y in [0,4): 16-col tile. WB3 fragments register-resident.
__global__ void edge_gemm_kernel(
    const float* __restrict__ e, const int* __restrict__ src,
    const int* __restrict__ dst, const float* __restrict__ WB3,
    const float* __restrict__ bB3, const float* __restrict__ B1h,
    const float* __restrict__ B2h, float* __restrict__ hat_eta,
    float* __restrict__ sum_sigma) {
  const int lane = threadIdx.x & 31;
  const int wave = threadIdx.x >> 5;
  const int half = lane >> 4;
  const int r    = lane & 15;
  const int col  = blockIdx.y * 16 + r;

  v2f bf[16];
#pragma unroll
  for (int ks = 0; ks < 16; ++ks) {
    const int k = ks * 4 + 2 * half;
    bf[ks].x = WB3[(k + 0) * D + col];
    bf[ks].y = WB3[(k + 1) * D + col];
  }
  const float bcol = bB3[col];

  const int tiles   = EE / 16;
  const int wstride = gridDim.x * 8;
  for (int tile = blockIdx.x * 8 + wave; tile < tiles; tile += wstride) {
    const int row0 = tile * 16;
    const float* erow = e + (size_t)(row0 + r) * D;
    v2f af[16];
#pragma unroll
    for (int ks = 0; ks < 16; ++ks) {
      const int k = ks * 4 + 2 * half;
      af[ks].x = erow[k]; af[ks].y = erow[k + 1];
    }
    v8f c = {};
#pragma unroll
    for (int ks = 0; ks < 16; ++ks)
      c = __builtin_amdgcn_wmma_f32_16x16x4_f32(false, af[ks], false, bf[ks],
                                                (short)0, c, false, false);
#pragma unroll
    for (int i = 0; i < 8; ++i) {
      const int er = row0 + i + 8 * half;
      const int s  = src[er];
      const int d2 = dst[er];
      const float v = c[i] + bcol + B1h[(size_t)s * D + col]
                                  + B2h[(size_t)d2 * D + col];
      hat_eta[(size_t)er * D + col] = v;
      const float sig = 1.0f / (1.0f + __expf(-v));
      unsafeAtomicAdd(&sum_sigma[(size_t)d2 * D + col], sig);
    }
  }
}

// -------- Normalize gates, scatter-add into h_new / p_new --------
__global__ void edge_aggregate_kernel(
    const float* __restrict__ hat_eta, const int* __restrict__ src,
    const int* __restrict__ dst, const float* __restrict__ sum_sigma,
    const float* __restrict__ A2hp, const float* __restrict__ C2p,
    float* __restrict__ h_acc, float* __restrict__ p_acc) {
  const size_t idx = (size_t)blockIdx.x * 256 + threadIdx.x;
  const int edge = (int)(idx >> 6);
  const int col  = (int)(idx & 63);
  const float v = hat_eta[idx];
  const float sig = 1.0f / (1.0f + __expf(-v));
  const int s = src[edge], d2 = dst[edge];
  const float eta = sig / (sum_sigma[(size_t)d2 * D + col] + 1e-6f);
  unsafeAtomicAdd(&h_acc[(size_t)d2 * D + col], eta * A2hp[(size_t)s * D + col]);
  unsafeAtomicAdd(&p_acc[(size_t)d2 * D + col], eta * C2p[(size_t)s * D + col]);
}

// -------- Per-column sum / sumsq for BatchNorm --------
__global__ void col_stats_kernel(const float* __restrict__ X, int rows,
                                 float* __restrict__ sum,
                                 float* __restrict__ sumsq) {
  const int col   = threadIdx.x & 63;
  const int rslot = threadIdx.x >> 6;          // 0..3
  const int row0  = blockIdx.x * 128 + rslot;
  float s = 0.0f, q = 0.0f;
  for (int i = 0; i < 32; ++i) {
    const int rr = row0 + i * 4;
    if (rr < rows) {
      const float x = X[(size_t)rr * D + col];
      s += x; q += x * x;
    }
  }
  unsafeAtomicAdd(&sum[col], s);
  unsafeAtomicAdd(&sumsq[col], q);
}

// -------- In-place BatchNorm + ReLU --------
__global__ void bn_relu_kernel(float* __restrict__ X, int rows,
                               const float* __restrict__ sum,
                               const float* __restrict__ sumsq,
                               const float* __restrict__ gamma,
                               const float* __restrict__ beta) {
  const size_t idx = (size_t)blockIdx.x * 256 + threadIdx.x;
  const int col = (int)(idx & 63);
  const float inv = 1.0f / (float)rows;
  const float mu = sum[col] * inv;
  const float var = sumsq[col] * inv - mu * mu;
  const float y = gamma[col] * (X[idx] - mu) * rsqrtf(var + 1e-5f) + beta[col];
  X[idx] = fmaxf(y, 0.0f);
}

__global__ void tanh_kernel(float* __restrict__ X) {
  const size_t idx = (size_t)blockIdx.x * 256 + threadIdx.x;
  X[idx] = tanhf(X[idx]);
}

extern "C" void kernel_launch(void* const* d_in, const int* in_sizes, int n_in,
                              void* d_out, int out_size, void* d_ws,
                              size_t ws_size, hipStream_t stream) {
  const float* h   = (const float*)d_in[0];
  const float* e   = (const float*)d_in[1];
  const float* p   = (const float*)d_in[2];
  const int*   src = (const int*)d_in[3];
  const int*   dst = (const int*)d_in[4];
  const float* WA1 = (const float*)d_in[5];  const float* bA1 = (const float*)d_in[6];
  const float* WA2 = (const float*)d_in[7];  const float* bA2 = (const float*)d_in[8];
  const float* WB1 = (const float*)d_in[9];  const float* bB1 = (const float*)d_in[10];
  const float* WB2 = (const float*)d_in[11]; const float* bB2 = (const float*)d_in[12];
  const float* WB3 = (const float*)d_in[13]; const float* bB3 = (const float*)d_in[14];
  const float* WC1 = (const float*)d_in[15]; const float* bC1 = (const float*)d_in[16];
  const float* WC2 = (const float*)d_in[17]; const float* bC2 = (const float*)d_in[18];
  const float* gamma_h = (const float*)d_in[19];
  const float* beta_h  = (const float*)d_in[20];
  const float* gamma_e = (const float*)d_in[21];
  const float* beta_e  = (const float*)d_in[22];

  float* out   = (float*)d_out;
  float* h_out = out;                               // [N, D]
  float* e_out = out + (size_t)NN * D;              // [E, D] (holds hat_eta first)
  float* p_out = e_out + (size_t)EE * D;            // [N, D]

  float* ws        = (float*)d_ws;
  float* A2hp      = ws;
  float* B1h       = ws + (size_t)NN * D;
  float* B2h       = ws + 2 * (size_t)NN * D;
  float* C2p       = ws + 3 * (size_t)NN * D;
  float* sum_sigma = ws + 4 * (size_t)NN * D;
  float* stats     = ws + 5 * (size_t)NN * D;       // sum_h, sq_h, sum_e, sq_e (4x64)

  // zero the accumulators (sum_sigma + 256 stat floats, contiguous)
  hipMemsetAsync(sum_sigma, 0, ((size_t)NN * D + 256) * sizeof(float), stream);

  // 24 slabs (6 outputs x 4 col tiles); each wave strides ~10 row tiles.
  node_gemm_kernel<<<dim3(40, 24), 256, 0, stream>>>(
      h, p, WA1, bA1, WA2, bA2, WB1, bB1, WB2, bB2, WC1, bC1, WC2, bC2,
      h_out /*A1h*/, A2hp, B1h, B2h, p_out /*C1p*/, C2p);

  // 4 col-tile slabs over 50000 edge tiles; ~10 tiles per wave.
  edge_gemm_kernel<<<dim3(625, 4), 256, 0, stream>>>(
      e, src, dst, WB3, bB3, B1h, B2h, e_out /*hat_eta*/, sum_sigma);

  edge_aggregate_kernel<<<(int)(((size_t)EE * D) / 256), 256, 0, stream>>>(
      e_out, src, dst, sum_sigma, A2hp, C2p, h_out, p_out);

  col_stats_kernel<<<(NN + 127) / 128, 256, 0, stream>>>(h_out, NN, stats,
                                                         stats + 64);
  col_stats_kernel<<<(EE + 127) / 128, 256, 0, stream>>>(e_out, EE, stats + 128,
                                                         stats + 192);

  bn_relu_kernel<<<(int)(((size_t)NN * D) / 256), 256, 0, stream>>>(
      h_out, NN, stats, stats + 64, gamma_h, beta_h);
  bn_relu_kernel<<<(int)(((size_t)EE * D) / 256), 256, 0, stream>>>(
      e_out, EE, stats + 128, stats + 192, gamma_e, beta_e);
  tanh_kernel<<<(int)(((size_t)NN * D) / 256), 256, 0, stream>>>(p_out);
}